// PyGHypergraphConvWrapper_7060926234637
// MI455X (gfx1250) — compile-verified
//
#include <hip/hip_runtime.h>
#include <hip/hip_bf16.h>

#define NUM_NODES 50000
#define NUM_EDGES 50000
#define NNZ       500000
#define D_IN      256
#define D_OUT     256

typedef float v2f __attribute__((ext_vector_type(2)));
typedef float v8f __attribute__((ext_vector_type(8)));

// ---------------------------------------------------------------------------
// Init: zero e_feat / D / Bc, seed out with the bias (out = sum + b later).
// ---------------------------------------------------------------------------
__global__ void hgc_init_kernel(float* __restrict__ e_feat,
                                float* __restrict__ Dn,
                                float* __restrict__ Bc,
                                float* __restrict__ out,
                                const float* __restrict__ b) {
    size_t idx = (size_t)blockIdx.x * blockDim.x + threadIdx.x;
    if (idx < (size_t)NUM_EDGES * D_OUT) e_feat[idx] = 0.0f;
    if (idx < (size_t)NUM_NODES * D_OUT) out[idx]    = b[idx & (D_OUT - 1)];
    if (idx < NUM_NODES) Dn[idx] = 0.0f;
    if (idx < NUM_EDGES) Bc[idx] = 0.0f;
}

// ---------------------------------------------------------------------------
// Degrees: D[n] += w[e], Bc[e] += 1 over all incidence entries.
// ---------------------------------------------------------------------------
__global__ void hgc_degree_kernel(const int* __restrict__ node_idx,
                                  const int* __restrict__ edge_idx,
                                  const float* __restrict__ w,
                                  float* __restrict__ Dn,
                                  float* __restrict__ Bc) {
    int i = blockIdx.x * blockDim.x + threadIdx.x;
    if (i < NNZ) {
        int n = node_idx[i];
        int e = edge_idx[i];
        atomicAdd(&Dn[n], w[e]);
        atomicAdd(&Bc[e], 1.0f);
    }
}

// ---------------------------------------------------------------------------
// xl = x @ W via V_WMMA_F32_16X16X4_F32. One 16x16 output tile per wave32,
// K loop over 256 in steps of 4 (64 chained WMMAs, fp32-exact accumulate).
// Grid: 6250 blocks x 256 threads = 50000 waves = (50000/16) x (256/16) tiles.
// ---------------------------------------------------------------------------
__global__ void hgc_gemm_wmma_kernel(const float* __restrict__ x,
                                     const float* __restrict__ W,
                                     float* __restrict__ xl) {
    const int tid   = threadIdx.x;
    const int wave  = tid >> 5;
    const int lane  = tid & 31;
    const int gwave = blockIdx.x * 8 + wave;        // 0..49999
    const int rowTile = gwave >> 4;                 // 0..3124
    const int colTile = gwave & 15;                 // 0..15
    const int row0 = rowTile * 16;
    const int col0 = colTile * 16;

    const int half = lane >> 4;                     // 0: K+{0,1}, 1: K+{2,3}
    const int l15  = lane & 15;

    // A: row (row0+l15) of x, starting at K offset half*2
    const float* arow = x + (size_t)(row0 + l15) * D_IN + half * 2;
    // B: column (col0+l15) of W, rows k + half*2 + {0,1}
    const float* bcol = W + (size_t)(half * 2) * D_OUT + (col0 + l15);

    v8f c = {};
#pragma unroll 4
    for (int k = 0; k < D_IN; k += 4) {
        v2f a, b;
        a.x = arow[k + 0];
        a.y = arow[k + 1];
        b.x = bcol[(size_t)(k + 0) * D_OUT];
        b.y = bcol[(size_t)(k + 1) * D_OUT];
        c = __builtin_amdgcn_wmma_f32_16x16x4_f32(
                /*neg_a=*/false, a, /*neg_b=*/false, b,
                /*c_mod=*/(short)0, c, /*reuse_a=*/false, /*reuse_b=*/false);
    }

    // C/D layout: VGPR r -> row r + 8*half, col l15
    float* orow = xl + (size_t)(row0 + half * 8) * D_OUT + col0 + l15;
#pragma unroll
    for (int r = 0; r < 8; ++r) {
        orow[(size_t)r * D_OUT] = c[r];
    }
}

// ---------------------------------------------------------------------------
// Stage 1 scatter: e_feat[e] += Binv[e] * xl[n].  64 threads per incidence
// entry, float4 gather + 4 global f32 atomics each. Working set lives in L2.
// ---------------------------------------------------------------------------
__global__ void hgc_scatter_n2e_kernel(const int* __restrict__ node_idx,
                                       const int* __restrict__ edge_idx,
                                       const float* __restrict__ Bc,
                                       const float* __restrict__ xl,
                                       float* __restrict__ e_feat) {
    size_t idx   = (size_t)blockIdx.x * blockDim.x + threadIdx.x;
    size_t entry = idx >> 6;                 // 64 threads per nnz entry
    int    f     = (int)(idx & 63) * 4;      // 4 features per thread
    if (entry >= NNZ) return;
    int n = node_idx[entry];
    int e = edge_idx[entry];
    float bc   = Bc[e];
    float binv = bc > 0.0f ? 1.0f / bc : 0.0f;
    const float4 v = *(const float4*)(xl + (size_t)n * D_OUT + f);
    float* dst = e_feat + (size_t)e * D_OUT + f;
    atomicAdd(dst + 0, binv * v.x);
    atomicAdd(dst + 1, binv * v.y);
    atomicAdd(dst + 2, binv * v.z);
    atomicAdd(dst + 3, binv * v.w);
}

// ---------------------------------------------------------------------------
// Stage 2 scatter: out[n] += Dinv[n] * e_feat[e]  (out pre-seeded with bias).
// ---------------------------------------------------------------------------
__global__ void hgc_scatter_e2n_kernel(const int* __restrict__ node_idx,
                                       const int* __restrict__ edge_idx,
                                       const float* __restrict__ Dn,
                                       const float* __restrict__ e_feat,
                                       float* __restrict__ out) {
    size_t idx   = (size_t)blockIdx.x * blockDim.x + threadIdx.x;
    size_t entry = idx >> 6;
    int    f     = (int)(idx & 63) * 4;
    if (entry >= NNZ) return;
    int n = node_idx[entry];
    int e = edge_idx[entry];
    float d    = Dn[n];
    float dinv = d > 0.0f ? 1.0f / d : 0.0f;
    const float4 v = *(const float4*)(e_feat + (size_t)e * D_OUT + f);
    float* dst = out + (size_t)n * D_OUT + f;
    atomicAdd(dst + 0, dinv * v.x);
    atomicAdd(dst + 1, dinv * v.y);
    atomicAdd(dst + 2, dinv * v.z);
    atomicAdd(dst + 3, dinv * v.w);
}

// ---------------------------------------------------------------------------
// Launcher. Inputs (setup_inputs order):
//   d_in[0] x                [N, 256] f32
//   d_in[1] hyperedge_index  [2, NNZ] int (row 0: node, row 1: edge)
//   d_in[2] hyperedge_weight [E] f32
//   d_in[3] W                [256, 256] f32
//   d_in[4] b                [256] f32
// ---------------------------------------------------------------------------
extern "C" void kernel_launch(void* const* d_in, const int* in_sizes, int n_in,
                              void* d_out, int out_size, void* d_ws, size_t ws_size,
                              hipStream_t stream) {
    const float* x        = (const float*)d_in[0];
    const int*   node_idx = (const int*)d_in[1];
    const int*   edge_idx = node_idx + NNZ;
    const float* w        = (const float*)d_in[2];
    const float* W        = (const float*)d_in[3];
    const float* b        = (const float*)d_in[4];
    float*       out      = (float*)d_out;

    // Workspace layout (floats): xl | e_feat | D | Bc   (~103 MB)
    float* xl     = (float*)d_ws;
    float* e_feat = xl + (size_t)NUM_NODES * D_OUT;
    float* Dn     = e_feat + (size_t)NUM_EDGES * D_OUT;
    float* Bc     = Dn + NUM_NODES;

    const int T = 256;

    // 1) init accumulators + bias seed
    {
        size_t total = (size_t)NUM_EDGES * D_OUT;   // == NUM_NODES*D_OUT
        int blocks = (int)((total + T - 1) / T);
        hgc_init_kernel<<<blocks, T, 0, stream>>>(e_feat, Dn, Bc, out, b);
    }
    // 2) degrees
    {
        int blocks = (NNZ + T - 1) / T;
        hgc_degree_kernel<<<blocks, T, 0, stream>>>(node_idx, edge_idx, w, Dn, Bc);
    }
    // 3) xl = x @ W  (WMMA)
    {
        int blocks = NUM_NODES / 8;                 // 8 waves/block, 1 tile/wave
        hgc_gemm_wmma_kernel<<<blocks, T, 0, stream>>>(x, W, xl);
    }
    // 4) node -> edge scatter
    {
        size_t total = (size_t)NNZ * 64;
        int blocks = (int)((total + T - 1) / T);
        hgc_scatter_n2e_kernel<<<blocks, T, 0, stream>>>(node_idx, edge_idx, Bc, xl, e_feat);
    }
    // 5) edge -> node scatter (+bias already seeded)
    {
        size_t total = (size_t)NNZ * 64;
        int blocks = (int)((total + T - 1) / T);
        hgc_scatter_e2n_kernel<<<blocks, T, 0, stream>>>(node_idx, edge_idx, Dn, e_feat, out);
    }
}